// EdgeConv_28303834480929
// MI455X (gfx1250) — compile-verified
//
#include <hip/hip_runtime.h>

// Problem constants (from reference): S=384, B=2, D=512, NHEAD=8 -> HG=16 head-groups, hd=64
#define HG   16
#define SP   384
#define HD   64
#define EPSF 1e-5f
#define SLOPEF 0.2f

typedef __attribute__((ext_vector_type(2))) float v2f;
typedef __attribute__((ext_vector_type(8))) float v8f;

// Workspace layout (float offsets)
#define OFF_A     0u         // 16*384*64 = 393216 ; overwritten with ap = s_k*A
#define OFF_C     393216u    // 393216 ; overwritten with cp = s_k*C + t_k
#define OFF_STAT  786432u    // 6 * 1024  (SC,SC2,SA,SA2,MAXA,MINA)[n*64+k]
#define OFF_SK    792576u    // 64
#define OFF_TK    792640u    // 64
#define OFF_SAEX  792704u    // 1024  s_k * extreme_j A[n,j,k]
#define OFF_W2P   793728u    // 64    0.4*W2
#define OFF_U     793792u    // 6144  0.6*sum_k W2_k*cp[n,i,k]
#define OFF_V     799936u    // 6144  0.6*sum_k W2_k*ap[n,j,k]
#define OFF_PART  806080u    // 2*1152 per-block (sum,sumsq) partials
#define OFF_ACC   808384u    // 2

__device__ __forceinline__ float lrelu(float z) { return z >= 0.f ? z : SLOPEF * z; }

// ---------------------------------------------------------------------------
// K1: A = xg * Wa^T, C = xg * (Wb-Wa)^T via V_WMMA_F32_16X16X4_F32.
// One wave per 16x16 output tile; A-fragment shared by both WMMAs per K-step.
// grid = 16 heads * 24 m-tiles * 4 n-tiles = 1536 waves
// ---------------------------------------------------------------------------
__global__ void __launch_bounds__(32)
gemm_ac_kernel(const float* __restrict__ x, const float* __restrict__ W1,
               float* __restrict__ Abuf, float* __restrict__ Cbuf) {
  int blk = blockIdx.x;
  int kt = blk & 3;           // k-tile (N dim), 4 tiles of 16 over 64
  int it = (blk >> 2) % 24;   // s-tile (M dim), 24 tiles of 16 over 384
  int h  = blk / 96;          // head-group 0..15
  int lane = threadIdx.x;     // wave32
  int half = lane >> 4;
  int lm   = lane & 15;

  // xg[h, s, d] = x[(s%192)*2 + h/8, s/192, (h%8)*64 + d], x is (384,2,512) row-major
  int s     = it * 16 + lm;
  int sorig = (s % 192) * 2 + (h >> 3);
  int borig = s / 192;
  const float* xrow  = x  + (size_t)sorig * 1024 + (size_t)borig * 512 + (h & 7) * 64;
  const float* w1row = W1 + (size_t)(kt * 16 + lm) * 128;

  v8f accA = {0.f,0.f,0.f,0.f,0.f,0.f,0.f,0.f};
  v8f accC = {0.f,0.f,0.f,0.f,0.f,0.f,0.f,0.f};
#pragma unroll
  for (int t = 0; t < 16; ++t) {
    int d = t * 4 + half * 2;          // A 16x4 layout: lanes0-15 K=0..1, lanes16-31 K=2..3
    v2f a  = { xrow[d],  xrow[d + 1] };
    v2f wa = { w1row[d], w1row[d + 1] };                       // B[kk,n] = W1[k0+n, d0+kk]
    v2f wc = { w1row[64 + d] - wa.x, w1row[64 + d + 1] - wa.y };
    accA = __builtin_amdgcn_wmma_f32_16x16x4_f32(false, a, false, wa, (short)0, accA, false, false);
    accC = __builtin_amdgcn_wmma_f32_16x16x4_f32(false, a, false, wc, (short)0, accC, false, false);
  }
  // C/D layout: VGPR r, lane -> row = r + 8*half, col = lane&15
  float* Adst = Abuf + (size_t)h * (SP * HD) + (size_t)(it * 16) * HD + kt * 16 + lm;
  float* Cdst = Cbuf + (size_t)h * (SP * HD) + (size_t)(it * 16) * HD + kt * 16 + lm;
#pragma unroll
  for (int r = 0; r < 8; ++r) {
    int row = r + 8 * half;
    Adst[row * HD] = accA[r];
    Cdst[row * HD] = accC[r];
  }
}

// ---------------------------------------------------------------------------
// K2: per-(n,k) reductions over s: sums/sumsqs of C and A, max/min of A.
// One wave per (n,k); 1024 blocks.
// ---------------------------------------------------------------------------
__global__ void __launch_bounds__(32)
stats_kernel(const float* __restrict__ Abuf, const float* __restrict__ Cbuf,
             float* __restrict__ stat) {
  int nk = blockIdx.x;             // n*64 + k
  int lane = threadIdx.x;
  const float* Ap = Abuf + (size_t)(nk >> 6) * (SP * HD) + (nk & 63);
  const float* Cp = Cbuf + (size_t)(nk >> 6) * (SP * HD) + (nk & 63);
  float sc = 0.f, sc2 = 0.f, sa = 0.f, sa2 = 0.f;
  float mx = -3.402823466e38f, mn = 3.402823466e38f;
  for (int s = lane; s < SP; s += 32) {
    float c = Cp[s * HD], a = Ap[s * HD];
    sc += c; sc2 += c * c; sa += a; sa2 += a * a;
    mx = fmaxf(mx, a); mn = fminf(mn, a);
  }
#pragma unroll
  for (int o = 16; o > 0; o >>= 1) {
    sc  += __shfl_xor(sc, o);  sc2 += __shfl_xor(sc2, o);
    sa  += __shfl_xor(sa, o);  sa2 += __shfl_xor(sa2, o);
    mx = fmaxf(mx, __shfl_xor(mx, o));
    mn = fminf(mn, __shfl_xor(mn, o));
  }
  if (lane == 0) {
    stat[0 * 1024 + nk] = sc;  stat[1 * 1024 + nk] = sc2;
    stat[2 * 1024 + nk] = sa;  stat[3 * 1024 + nk] = sa2;
    stat[4 * 1024 + nk] = mx;  stat[5 * 1024 + nk] = mn;
  }
}

// ---------------------------------------------------------------------------
// K3: closed-form BN1 stats.  mean_k = 384*(ΣC+ΣA)/N ; E[pre²] uses cross term.
// ---------------------------------------------------------------------------
__global__ void __launch_bounds__(64)
combine_kernel(const float* __restrict__ stat,
               const float* __restrict__ gamma1, const float* __restrict__ beta1,
               const float* __restrict__ W2,
               float* __restrict__ sk, float* __restrict__ tk,
               float* __restrict__ saext, float* __restrict__ w2p,
               float* __restrict__ acc) {
  int k = threadIdx.x;  // 0..63
  float tsc = 0.f, tsc2 = 0.f, tsa = 0.f, tsa2 = 0.f, cross = 0.f;
#pragma unroll
  for (int n = 0; n < HG; ++n) {
    float sc  = stat[0 * 1024 + n * 64 + k];
    float sc2 = stat[1 * 1024 + n * 64 + k];
    float sa  = stat[2 * 1024 + n * 64 + k];
    float sa2 = stat[3 * 1024 + n * 64 + k];
    tsc += sc; tsc2 += sc2; tsa += sa; tsa2 += sa2; cross += sc * sa;
  }
  const float Ntot = (float)HG * SP * SP;                 // 2359296
  float mean  = (float)SP * (tsc + tsa) / Ntot;
  float sumsq = (float)SP * (tsc2 + tsa2) + 2.f * cross;
  float var   = sumsq / Ntot - mean * mean;
  float s = gamma1[k] * rsqrtf(var + EPSF);
  float t = beta1[k] - s * mean;
  sk[k] = s; tk[k] = t;
#pragma unroll
  for (int n = 0; n < HG; ++n) {
    float mx = stat[4 * 1024 + n * 64 + k];
    float mn = stat[5 * 1024 + n * 64 + k];
    saext[n * 64 + k] = s * (s >= 0.f ? mx : mn);          // s_k * argextreme_j A
  }
  w2p[k] = 0.4f * W2[k];                                   // 0.4 = (1-slope)/2
  if (k < 2) acc[k] = 0.f;
}

// ---------------------------------------------------------------------------
// K4: in-place cp = s_k*C + t_k, ap = s_k*A; x_out; U' = 0.6*Σ W2_k*cp, V' likewise.
// One block of 64 threads (2 waves) per (n,s).
// ---------------------------------------------------------------------------
__global__ void __launch_bounds__(64)
prep_kernel(float* __restrict__ Abuf, float* __restrict__ Cbuf,
            const float* __restrict__ sk, const float* __restrict__ tk,
            const float* __restrict__ saext, const float* __restrict__ W2,
            float* __restrict__ Uv, float* __restrict__ Vv,
            float* __restrict__ xout) {
  __shared__ float red[4];
  int bs = blockIdx.x;               // n*384 + s
  int n = bs / SP;
  int k = threadIdx.x;
  size_t idx = (size_t)bs * HD + k;
  float s = sk[k], t = tk[k];
  float c = Cbuf[idx], a = Abuf[idx];
  float cp = s * c + t;
  float ap = s * a;
  Cbuf[idx] = cp;
  Abuf[idx] = ap;
  xout[idx] = lrelu(cp + saext[n * 64 + k]);               // max_j pulled through monotone lrelu
  float w = W2[k];
  float u = w * cp, v = w * ap;
#pragma unroll
  for (int o = 16; o > 0; o >>= 1) { u += __shfl_xor(u, o); v += __shfl_xor(v, o); }
  int wid = k >> 5;
  if ((k & 31) == 0) { red[wid * 2] = u; red[wid * 2 + 1] = v; }
  __syncthreads();
  if (k == 0) {
    Uv[bs] = 0.6f * (red[0] + red[2]);                     // 0.6 = (1+slope)/2
    Vv[bs] = 0.6f * (red[1] + red[3]);
  }
}

// ---------------------------------------------------------------------------
// K5: pairwise y[n,i,j] = U'_i + V'_j + Σ_k w2p_k*|cp_ik + ap_jk|.
// Tile 32i x 64j per 256-thread block; LDS padded to 65 (bank-conflict free).
// Per-block (sum,sumsq) partials written for deterministic BN2 reduction.
// grid = (6 j-tiles, 12 i-tiles, 16 heads)
// ---------------------------------------------------------------------------
__global__ void __launch_bounds__(256)
pair_kernel(const float* __restrict__ Abuf /*ap*/, const float* __restrict__ Cbuf /*cp*/,
            const float* __restrict__ w2p,
            const float* __restrict__ Uv, const float* __restrict__ Vv,
            float* __restrict__ e, float* __restrict__ part) {
  __shared__ float cs[32][65];
  __shared__ float as[64][65];
  __shared__ float ws2[64];
  __shared__ float redS[8], redQ[8];
  int n = blockIdx.z, it = blockIdx.y, jt = blockIdx.x;
  int i0 = it * 32, j0 = jt * 64;
  int t = threadIdx.x;
  const float* cpB = Cbuf + (size_t)n * (SP * HD) + (size_t)i0 * HD;
  const float* apB = Abuf + (size_t)n * (SP * HD) + (size_t)j0 * HD;
  for (int i = t; i < 32 * 64; i += 256) cs[i >> 6][i & 63] = cpB[i];
  for (int i = t; i < 64 * 64; i += 256) as[i >> 6][i & 63] = apB[i];
  if (t < 64) ws2[t] = w2p[t];
  __syncthreads();

  int tx = t & 15;           // j-group of 4
  int ty = t >> 4;           // i-pair
  int il = ty * 2, jl = tx * 4;
  float a00 = 0.f, a01 = 0.f, a02 = 0.f, a03 = 0.f;
  float a10 = 0.f, a11 = 0.f, a12 = 0.f, a13 = 0.f;
#pragma unroll 8
  for (int k = 0; k < 64; ++k) {
    float w  = ws2[k];
    float c0 = cs[il][k],     c1 = cs[il + 1][k];
    float b0 = as[jl][k],     b1 = as[jl + 1][k];
    float b2 = as[jl + 2][k], b3 = as[jl + 3][k];
    a00 += w * fabsf(c0 + b0);  a01 += w * fabsf(c0 + b1);
    a02 += w * fabsf(c0 + b2);  a03 += w * fabsf(c0 + b3);
    a10 += w * fabsf(c1 + b0);  a11 += w * fabsf(c1 + b1);
    a12 += w * fabsf(c1 + b2);  a13 += w * fabsf(c1 + b3);
  }
  int bi = n * SP + i0 + il, bj = n * SP + j0 + jl;
  float u0 = Uv[bi], u1 = Uv[bi + 1];
  float v0 = Vv[bj], v1 = Vv[bj + 1], v2 = Vv[bj + 2], v3 = Vv[bj + 3];
  float y00 = a00 + u0 + v0, y01 = a01 + u0 + v1, y02 = a02 + u0 + v2, y03 = a03 + u0 + v3;
  float y10 = a10 + u1 + v0, y11 = a11 + u1 + v1, y12 = a12 + u1 + v2, y13 = a13 + u1 + v3;
  float* er = e + (size_t)n * (SP * SP) + (size_t)(i0 + il) * SP + (j0 + jl);
  er[0] = y00; er[1] = y01; er[2] = y02; er[3] = y03;
  er[SP + 0] = y10; er[SP + 1] = y11; er[SP + 2] = y12; er[SP + 3] = y13;

  float s = y00 + y01 + y02 + y03 + y10 + y11 + y12 + y13;
  float q = y00*y00 + y01*y01 + y02*y02 + y03*y03 + y10*y10 + y11*y11 + y12*y12 + y13*y13;
#pragma unroll
  for (int o = 16; o > 0; o >>= 1) { s += __shfl_xor(s, o); q += __shfl_xor(q, o); }
  int wid = t >> 5;
  if ((t & 31) == 0) { redS[wid] = s; redQ[wid] = q; }
  __syncthreads();
  if (t == 0) {
    float ts = 0.f, tq = 0.f;
#pragma unroll
    for (int w2 = 0; w2 < 8; ++w2) { ts += redS[w2]; tq += redQ[w2]; }
    int bflat = (n * 12 + it) * 6 + jt;          // 0..1151
    part[2 * bflat] = ts; part[2 * bflat + 1] = tq;
  }
}

// ---------------------------------------------------------------------------
// K6: deterministic fixed-order reduction of the 1152 block partials.
// ---------------------------------------------------------------------------
__global__ void __launch_bounds__(256)
reduce_part_kernel(const float* __restrict__ part, float* __restrict__ acc) {
  __shared__ float rs[256], rq[256];
  int t = threadIdx.x;
  float s = 0.f, q = 0.f;
  for (int i = t; i < 1152; i += 256) { s += part[2 * i]; q += part[2 * i + 1]; }
  rs[t] = s; rq[t] = q;
  __syncthreads();
  for (int o = 128; o > 0; o >>= 1) {
    if (t < o) { rs[t] += rs[t + o]; rq[t] += rq[t + o]; }
    __syncthreads();
  }
  if (t == 0) { acc[0] = rs[0]; acc[1] = rq[0]; }
}

// ---------------------------------------------------------------------------
// K7: BN2 + lrelu in place over the e region of d_out.
// ---------------------------------------------------------------------------
__global__ void __launch_bounds__(256)
final_kernel(float* __restrict__ e, const float* __restrict__ acc,
             const float* __restrict__ gamma2, const float* __restrict__ beta2) {
  int idx = blockIdx.x * 256 + threadIdx.x;   // grid sized exactly
  const float Ntot = (float)HG * SP * SP;
  float mean = acc[0] / Ntot;
  float var  = acc[1] / Ntot - mean * mean;
  float inv  = rsqrtf(var + EPSF) * gamma2[0];
  float b    = beta2[0];
  float y  = e[idx];
  float yn = (y - mean) * inv + b;
  e[idx] = lrelu(yn);
}

extern "C" void kernel_launch(void* const* d_in, const int* in_sizes, int n_in,
                              void* d_out, int out_size, void* d_ws, size_t ws_size,
                              hipStream_t stream) {
  (void)in_sizes; (void)n_in; (void)out_size; (void)ws_size;
  const float* x      = (const float*)d_in[0];
  const float* W1     = (const float*)d_in[1];
  const float* gamma1 = (const float*)d_in[2];
  const float* beta1  = (const float*)d_in[3];
  const float* W2     = (const float*)d_in[4];
  const float* gamma2 = (const float*)d_in[5];
  const float* beta2  = (const float*)d_in[6];
  float* out = (float*)d_out;
  float* ws  = (float*)d_ws;

  float* A     = ws + OFF_A;
  float* C     = ws + OFF_C;
  float* stat  = ws + OFF_STAT;
  float* sk    = ws + OFF_SK;
  float* tk    = ws + OFF_TK;
  float* saext = ws + OFF_SAEX;
  float* w2p   = ws + OFF_W2P;
  float* Uv    = ws + OFF_U;
  float* Vv    = ws + OFF_V;
  float* part  = ws + OFF_PART;
  float* acc   = ws + OFF_ACC;

  float* xout = out;                       // (16,384,64) = 393216 floats
  float* e    = out + (size_t)HG * SP * HD;  // (16,384,384) = 2359296 floats

  gemm_ac_kernel<<<HG * 24 * 4, 32, 0, stream>>>(x, W1, A, C);
  stats_kernel<<<HG * HD, 32, 0, stream>>>(A, C, stat);
  combine_kernel<<<1, 64, 0, stream>>>(stat, gamma1, beta1, W2, sk, tk, saext, w2p, acc);
  prep_kernel<<<HG * SP, 64, 0, stream>>>(A, C, sk, tk, saext, W2, Uv, Vv, xout);
  pair_kernel<<<dim3(6, 12, HG), 256, 0, stream>>>(A, C, w2p, Uv, Vv, e, part);
  reduce_part_kernel<<<1, 256, 0, stream>>>(part, acc);
  final_kernel<<<(HG * SP * SP) / 256, 256, 0, stream>>>(e, acc, gamma2, beta2);
}